// SSM_49667001811249
// MI455X (gfx1250) — compile-verified
//
#include <hip/hip_runtime.h>
#include <hip/hip_bf16.h>
#include <math.h>

typedef __attribute__((ext_vector_type(2))) float v2f;
typedef __attribute__((ext_vector_type(4))) float v4f;
typedef __attribute__((ext_vector_type(8))) float v8f;

#define B_DIM 8192
#define D_DIM 4096
#define ROWS_PER_BLK 16          // one WMMA M-tile per block
#define K1_WAVES 8
#define K_SLICE (D_DIM / K1_WAVES)   // 512 K per wave -> 128 wmma each
#define CHUNK 128                // scan chunk per lane (32 lanes * 128 = 4096)

// ---------------------------------------------------------------------------
// Kernel 1: per-row dot products via V_WMMA_F32_16X16X4_F32, then coefficients
//   coef[row] = { a = softplus(u.dw + db + bias1) * (u.w1),  b = u.w2,  c = u.w3 }
// ---------------------------------------------------------------------------
__global__ __launch_bounds__(256) void ssm_coef_wmma(
    const float* __restrict__ u,  const float* __restrict__ w1,
    const float* __restrict__ w2, const float* __restrict__ w3,
    const float* __restrict__ dw, const float* __restrict__ bias1,
    const float* __restrict__ delta_b, float* __restrict__ coef)
{
    __shared__ float red[K1_WAVES * 16 * 16];   // per-wave C tiles, 8 KB

    const int tid  = threadIdx.x;
    const int wave = tid >> 5;
    const int lane = tid & 31;
    const int rowBase = blockIdx.x * ROWS_PER_BLK;

    const int m    = lane & 15;          // A row / B column index for this lane
    const int koff = (lane >> 4) * 2;    // lanes 0-15 -> K{0,1}; 16-31 -> K{2,3}

    // B matrix columns: N=0..3 hold w1,w2,w3,delta_w; N>=4 are zero.
    // Lanes n>=4 load from a valid pointer (dw) and are zeroed by `msk` —
    // branchless, so the WMMA loop has no EXEC save/restore serialization.
    const float* wsel = (m == 0) ? w1 : (m == 1) ? w2 : (m == 2) ? w3 : dw;
    const float  msk  = (m < 4) ? 1.0f : 0.0f;

    const float* arow = u + (size_t)(rowBase + m) * D_DIM + koff;
    const float* brow = wsel + koff;

    v8f acc = {};
    const int k0 = wave * K_SLICE;
    #pragma unroll 4
    for (int k = k0; k < k0 + K_SLICE; k += 4) {
        // A frag: 16x4 f32 tile of u (layout per ISA 7.12.2), 8B-aligned
        v2f afr = *(const v2f*)(arow + k);
        // B frag: 4x16 tile, columns >=4 masked to zero (no branch)
        v2f bfr = *(const v2f*)(brow + k);
        bfr.x *= msk;
        bfr.y *= msk;
        acc = __builtin_amdgcn_wmma_f32_16x16x4_f32(
            false, afr, false, bfr, (short)0, acc, false, false);
    }

    // C/D layout: VGPR v holds M = v + 8*(lane/16), N = lane%16
    #pragma unroll
    for (int v = 0; v < 8; ++v) {
        const int M = v + ((lane >> 4) << 3);
        red[(wave * 16 + M) * 16 + m] = acc[v];
    }
    __syncthreads();

    if (tid < 16) {
        float d1 = 0.f, d2 = 0.f, d3 = 0.f, dd = 0.f;
        #pragma unroll
        for (int w = 0; w < K1_WAVES; ++w) {
            const float* p = &red[(w * 16 + tid) * 16];
            d1 += p[0]; d2 += p[1]; d3 += p[2]; dd += p[3];
        }
        // select = softplus(u.dw + delta_b + bias1), numerically stable
        const float z  = dd + delta_b[0] + bias1[0];
        const float sp = fmaxf(z, 0.f) + log1pf(expf(-fabsf(z)));
        const int row = rowBase + tid;
        coef[row * 4 + 0] = sp * d1;   // a
        coef[row * 4 + 1] = d2;        // b
        coef[row * 4 + 2] = d3;        // c
        coef[row * 4 + 3] = 0.f;
    }
}

// ---------------------------------------------------------------------------
// Kernel 2: per-row first-order linear scan, wave-parallel via affine
// composition:  x_t = a*x_{t-1} + b*u_t  (x_{-1}=0),
//               y_t = c*x_t + Dc*u_t  (y_0 = x_0).
// One wave per row; lane j owns time chunk [128j, 128j+128).
// y stored non-temporally so u stays resident in the 192MB L2 (read from L2,
// not HBM, after kernel 1 streamed it).
// ---------------------------------------------------------------------------
__global__ __launch_bounds__(256) void ssm_scan(
    const float* __restrict__ u, const float* __restrict__ coef,
    const float* __restrict__ bias2, float* __restrict__ y)
{
    const int lane = threadIdx.x & 31;
    const int wave = threadIdx.x >> 5;
    const int row  = blockIdx.x * 8 + wave;

    const float a  = coef[row * 4 + 0];
    const float bc = coef[row * 4 + 1];
    const float cc = coef[row * 4 + 2];
    const float Dc = bias2[0];

    const float* up = u + (size_t)row * D_DIM + lane * CHUNK;
    float*       yp = y + (size_t)row * D_DIM + lane * CHUNK;

    // ---- pass 1: local scan with zero carry-in -> residual r = R(lane)
    float x = 0.f;
    #pragma unroll 4
    for (int t = 0; t < CHUNK; t += 4) {
        v4f uv = *(const v4f*)(up + t);
        #pragma unroll
        for (int j = 0; j < 4; ++j)
            x = fmaf(a, x, bc * uv[j]);
    }

    // A = a^128 (same for every lane)
    float A = a;
    #pragma unroll
    for (int i = 0; i < 7; ++i) A *= A;

    // ---- inclusive scan of affine maps f(x) = Ac*x + R across 32 lanes
    float R = x, Ac = A;
    #pragma unroll
    for (int d = 1; d < 32; d <<= 1) {
        const float Rp = __shfl_up(R, d, 32);
        const float Ap = __shfl_up(Ac, d, 32);
        if (lane >= d) { R = fmaf(Ac, Rp, R); Ac *= Ap; }
    }
    float x_in = __shfl_up(R, 1, 32);   // exclusive carry-in
    if (lane == 0) x_in = 0.f;

    // ---- pass 2: recompute with true carry, emit y (streaming stores)
    x = x_in;
    const int gt0 = lane * CHUNK;       // global time of this lane's first elem
    #pragma unroll 4
    for (int t = 0; t < CHUNK; t += 4) {
        v4f uv = *(const v4f*)(up + t);
        v4f yv;
        #pragma unroll
        for (int j = 0; j < 4; ++j) {
            x = fmaf(a, x, bc * uv[j]);
            float yj = fmaf(cc, x, Dc * uv[j]);
            if (gt0 + t + j == 0) yj = x;   // y[:,0] = x0 (no C/D term)
            yv[j] = yj;
        }
        __builtin_nontemporal_store(yv, (v4f*)(yp + t));
    }
}

// ---------------------------------------------------------------------------
extern "C" void kernel_launch(void* const* d_in, const int* in_sizes, int n_in,
                              void* d_out, int out_size, void* d_ws, size_t ws_size,
                              hipStream_t stream) {
    const float* u     = (const float*)d_in[0];
    const float* w1    = (const float*)d_in[1];
    const float* w2    = (const float*)d_in[2];
    const float* w3    = (const float*)d_in[3];
    const float* bias1 = (const float*)d_in[4];
    const float* bias2 = (const float*)d_in[5];
    const float* dw    = (const float*)d_in[6];
    const float* db    = (const float*)d_in[7];
    float* y    = (float*)d_out;
    float* coef = (float*)d_ws;   // 8192 * 4 floats = 128 KB

    ssm_coef_wmma<<<B_DIM / ROWS_PER_BLK, 256, 0, stream>>>(
        u, w1, w2, w3, dw, bias1, db, coef);
    ssm_scan<<<B_DIM / 8, 256, 0, stream>>>(u, coef, bias2, y);
}